// GNNPathPolicy_89670327206300
// MI455X (gfx1250) — compile-verified
//
#include <hip/hip_runtime.h>
#include <hip/hip_bf16.h>

typedef float v2f __attribute__((ext_vector_type(2)));
typedef float v8f __attribute__((ext_vector_type(8)));

#define NF 64   // node feature dim F == hidden H
#define FE 16   // edge feature dim
#define PLEN 20
#define KPATH 32
#define NACT 256

// ---------------------------------------------------------------------------
// Zero-fill
// ---------------------------------------------------------------------------
__global__ void fill_zero_kernel(float* __restrict__ p, int n) {
    int i = blockIdx.x * blockDim.x + threadIdx.x;
    if (i < n) p[i] = 0.0f;
}

// ---------------------------------------------------------------------------
// In-degree counts (real edges only): cnt[dst] += 1
// ---------------------------------------------------------------------------
__global__ void count_kernel(const int* __restrict__ dst, float* __restrict__ cnt, int E) {
    int i = blockIdx.x * blockDim.x + threadIdx.x;
    if (i < E) atomicAdd(&cnt[dst[i]], 1.0f);
}

// ---------------------------------------------------------------------------
// C[M x 64] = A[M x 64] @ B[64 x 64] (+ bias), row-major, f32 WMMA.
// Block = 256 threads = 8 waves; each wave owns one 16x16 C tile of a
// 32(rows) x 64(cols) block slab. K=64 -> 16 chained v_wmma_f32_16x16x4_f32.
// A/B staged in LDS. EXEC is all-ones for every WMMA (guards only on ld/st).
// ---------------------------------------------------------------------------
__global__ __launch_bounds__(256) void gemm_n64_k64_wmma(
    const float* __restrict__ A, const float* __restrict__ B,
    const float* __restrict__ bias, float* __restrict__ C, int M)
{
    __shared__ float sA[32 * 64];   //  8 KB
    __shared__ float sB[64 * 64];   // 16 KB
    const int tid  = threadIdx.x;
    const int wave = tid >> 5;
    const int lane = tid & 31;
    const int rowBase = blockIdx.x * 32;

    // cooperative, coalesced tile loads
    for (int i = tid; i < 32 * 64; i += 256) {
        int r = rowBase + (i >> 6);
        sA[i] = (r < M) ? A[r * 64 + (i & 63)] : 0.0f;
    }
    for (int i = tid; i < 64 * 64; i += 256) sB[i] = B[i];
    __syncthreads();

    const int tr   = (wave >> 2) * 16;  // 0 or 16 (row tile inside slab)
    const int tc   = (wave & 3) * 16;   // 0,16,32,48 (col tile)
    const int half = lane >> 4;         // 0: lanes 0-15, 1: lanes 16-31
    const int l15  = lane & 15;

    v8f c = {};
#pragma unroll
    for (int k = 0; k < 64; k += 4) {
        const int kk = k + 2 * half;
        v2f a, b;
        // A 16x4 fragment: lane holds row (lane&15), K = kk, kk+1
        a[0] = sA[(tr + l15) * 64 + kk];
        a[1] = sA[(tr + l15) * 64 + kk + 1];
        // B 4x16 fragment: lane holds col (lane&15), K = kk, kk+1
        b[0] = sB[kk * 64 + tc + l15];
        b[1] = sB[(kk + 1) * 64 + tc + l15];
        c = __builtin_amdgcn_wmma_f32_16x16x4_f32(false, a, false, b,
                                                  (short)0, c, false, false);
    }

    // C 16x16: VGPR j -> row = j + 8*half, col = lane&15
#pragma unroll
    for (int j = 0; j < 8; ++j) {
        int r   = rowBase + tr + j + 8 * half;
        int col = tc + l15;
        if (r < M) {
            float v = c[j];
            if (bias) v += bias[col];
            C[r * 64 + col] = v;
        }
    }
}

// ---------------------------------------------------------------------------
// Per-layer folded edge-attention vector:
//   v_e[h] = sum_j lin_e[h][j] * a_e[j]        (H)
//   u[t]   = sum_h W_eenc[t][h] * v_e[h]       (FE)   -> uc[0..15]
//   c      = sum_h b_eenc[h]   * v_e[h]               -> uc[16]
// ---------------------------------------------------------------------------
__global__ void ve_u_kernel(const float* __restrict__ lin_e, const float* __restrict__ a_e,
                            const float* __restrict__ W_eenc, const float* __restrict__ b_eenc,
                            float* __restrict__ uc)
{
    __shared__ float ve[64];
    int t = threadIdx.x;                // 64 threads
    float s = 0.0f;
    for (int j = 0; j < 64; ++j) s += lin_e[t * 64 + j] * a_e[j];
    ve[t] = s;
    __syncthreads();
    if (t < FE) {
        float us = 0.0f;
        for (int h = 0; h < 64; ++h) us += W_eenc[t * 64 + h] * ve[h];
        uc[t] = us;
    }
    if (t == 63) {
        float cc = 0.0f;
        for (int h = 0; h < 64; ++h) cc += b_eenc[h] * ve[h];
        uc[16] = cc;
    }
}

// ---------------------------------------------------------------------------
// al_s[n] = xs[n] . a_src ; al_d[n] = xs[n] . a_dst
// ---------------------------------------------------------------------------
__global__ void alsd_kernel(const float* __restrict__ xs,
                            const float* __restrict__ a_src, const float* __restrict__ a_dst,
                            float* __restrict__ al_s, float* __restrict__ al_d, int N)
{
    int n = blockIdx.x * blockDim.x + threadIdx.x;
    if (n >= N) return;
    const float4* row = (const float4*)(xs + (long)n * 64);
    float ss = 0.0f, sd = 0.0f;
#pragma unroll
    for (int q = 0; q < 16; ++q) {
        float4 v = row[q];
        ss += v.x * a_src[4*q] + v.y * a_src[4*q+1] + v.z * a_src[4*q+2] + v.w * a_src[4*q+3];
        sd += v.x * a_dst[4*q] + v.y * a_dst[4*q+1] + v.z * a_dst[4*q+2] + v.w * a_dst[4*q+3];
    }
    al_s[n] = ss;
    al_d[n] = sd;
}

// ---------------------------------------------------------------------------
// al_e[i] = edge_attr[i] . u + c  (real edges) ; sum_ale[dst] += al_e[i]
// ---------------------------------------------------------------------------
__global__ void ale_kernel(const float* __restrict__ edge_attr, const int* __restrict__ dst,
                           const float* __restrict__ uc,
                           float* __restrict__ al_e, float* __restrict__ sum_ale, int E)
{
    int i = blockIdx.x * blockDim.x + threadIdx.x;
    if (i >= E) return;
    const float4* row = (const float4*)(edge_attr + (long)i * FE);
    float s = uc[16];
#pragma unroll
    for (int q = 0; q < 4; ++q) {
        float4 v = row[q];
        s += v.x * uc[4*q] + v.y * uc[4*q+1] + v.z * uc[4*q+2] + v.w * uc[4*q+3];
    }
    al_e[i] = s;
    atomicAdd(&sum_ale[dst[i]], s);
}

// loopale[n] = sum_ale[n] / max(cnt[n], 1)
__global__ void loopale_kernel(const float* __restrict__ sum_ale, const float* __restrict__ cnt,
                               float* __restrict__ loopale, int N)
{
    int n = blockIdx.x * blockDim.x + threadIdx.x;
    if (n < N) loopale[n] = sum_ale[n] / fmaxf(cnt[n], 1.0f);
}

// ---------------------------------------------------------------------------
// Softmax-weighted aggregation (segment softmax is shift-invariant, so no
// segment-max pass is needed). One wave per (real|self-loop) edge:
//   p = exp(leaky_relu(al_s[src] + al_d[dst] + al_e))
//   denom[dst] += p ; acc[dst][:] += p * xs[src][:]   (2 features / lane)
// ---------------------------------------------------------------------------
__global__ __launch_bounds__(256) void edge_agg_kernel(
    const int* __restrict__ src, const int* __restrict__ dst,
    const float* __restrict__ al_s, const float* __restrict__ al_d,
    const float* __restrict__ al_e, const float* __restrict__ loopale,
    const float* __restrict__ xs,
    float* __restrict__ acc, float* __restrict__ denom, int E, int N)
{
    int e = blockIdx.x * 8 + (threadIdx.x >> 5);
    int lane = threadIdx.x & 31;
    if (e >= E + N) return;
    int s, d; float ale;
    if (e < E) { s = src[e]; d = dst[e]; ale = al_e[e]; }
    else       { s = e - E;  d = s;      ale = loopale[s]; }
    float logit = al_s[s] + al_d[d] + ale;
    logit = logit > 0.0f ? logit : 0.2f * logit;     // leaky_relu(0.2)
    float p = expf(logit);
    if (lane == 0) atomicAdd(&denom[d], p);
    float2 xv = *(const float2*)(xs + (long)s * 64 + 2 * lane);
    atomicAdd(&acc[(long)d * 64 + 2 * lane],     p * xv.x);
    atomicAdd(&acc[(long)d * 64 + 2 * lane + 1], p * xv.y);
}

// h[n][f] = relu(acc[n][f] / denom[n] + bias[f])   (denom > 0: self-loop)
__global__ void finalize_kernel(const float* __restrict__ acc, const float* __restrict__ denom,
                                const float* __restrict__ bias, float* __restrict__ h, int N)
{
    int i = blockIdx.x * blockDim.x + threadIdx.x;
    if (i >= N * 64) return;
    int n = i >> 6, f = i & 63;
    float v = acc[i] / denom[n] + bias[f];
    h[i] = fmaxf(v, 0.0f);
}

// ---------------------------------------------------------------------------
// Path encodings. 64 threads (one per feature).
//   s[f]    = sum_i w_i (h[path_i][f] + pos_emb[i][f]) / sum(w)
//   emb     = s @ W_comb + b_comb ; coll = mean_k(s_k) @ W_comb + b_comb
// ---------------------------------------------------------------------------
__global__ void path_kernel(const float* __restrict__ h,
                            const int* __restrict__ cur_path, const int* __restrict__ coll_paths,
                            const float* __restrict__ pos_emb,
                            const float* __restrict__ W_comb, const float* __restrict__ b_comb,
                            float* __restrict__ path_emb, float* __restrict__ coll_emb)
{
    __shared__ float scur[64], sbar[64];
    int f = threadIdx.x;
    float Wsum = 0.0f, s = 0.0f;
    for (int i = 0; i < PLEN; ++i) {
        float w = 0.5f + 0.5f * (float)i / (float)(PLEN - 1);   // linspace(0.5,1,20)
        Wsum += w;
        s += w * (h[(long)cur_path[i] * 64 + f] + pos_emb[i * 64 + f]);
    }
    scur[f] = s / Wsum;
    float sb = 0.0f;
    for (int k = 0; k < KPATH; ++k) {
        float sk = 0.0f;
        for (int i = 0; i < PLEN; ++i) {
            float w = 0.5f + 0.5f * (float)i / (float)(PLEN - 1);
            sk += w * (h[(long)coll_paths[k * PLEN + i] * 64 + f] + pos_emb[i * 64 + f]);
        }
        sb += sk / Wsum;
    }
    sbar[f] = sb / (float)KPATH;
    __syncthreads();
    float pe = b_comb[f], ce = b_comb[f];
    for (int t = 0; t < 64; ++t) {
        float wv = W_comb[t * 64 + f];
        pe += scur[t] * wv;
        ce += sbar[t] * wv;
    }
    path_emb[f] = pe;
    coll_emb[f] = ce;
}

// ---------------------------------------------------------------------------
// Scoring MLP + softmax over the 256 actions. One block, thread = action.
// feat = [cur, act, path_emb, coll_emb] (4H=256)
// score = relu(feat @ W_s1 + b_s1) @ W_s2 + b_s2 ; out = softmax(score)
// ---------------------------------------------------------------------------
__global__ __launch_bounds__(256) void score_kernel(
    const float* __restrict__ h, const int* __restrict__ valid_actions,
    const int* __restrict__ current_node,
    const float* __restrict__ path_emb, const float* __restrict__ coll_emb,
    const float* __restrict__ W_s1, const float* __restrict__ b_s1,
    const float* __restrict__ W_s2, const float* __restrict__ b_s2,
    float* __restrict__ out)
{
    __shared__ float cur[64], pe[64], ce[64];
    __shared__ float red[256];
    int t = threadIdx.x;
    if (t < 64) {
        cur[t] = h[(long)current_node[0] * 64 + t];
        pe[t]  = path_emb[t];
        ce[t]  = coll_emb[t];
    }
    __syncthreads();

    const float* act = h + (long)valid_actions[t] * 64;
    float score = b_s2[0];
    for (int j = 0; j < 64; ++j) {
        float z = b_s1[j];
        for (int i = 0; i < 64; ++i) z += cur[i] * W_s1[i * 64 + j];
        for (int i = 0; i < 64; ++i) z += act[i] * W_s1[(64 + i) * 64 + j];
        for (int i = 0; i < 64; ++i) z += pe[i]  * W_s1[(128 + i) * 64 + j];
        for (int i = 0; i < 64; ++i) z += ce[i]  * W_s1[(192 + i) * 64 + j];
        score += fmaxf(z, 0.0f) * W_s2[j];
    }

    red[t] = score; __syncthreads();
    for (int off = 128; off > 0; off >>= 1) {
        if (t < off) red[t] = fmaxf(red[t], red[t + off]);
        __syncthreads();
    }
    float m = red[0]; __syncthreads();
    float ex = expf(score - m);
    red[t] = ex; __syncthreads();
    for (int off = 128; off > 0; off >>= 1) {
        if (t < off) red[t] += red[t + off];
        __syncthreads();
    }
    out[t] = ex / red[0];
}

// ---------------------------------------------------------------------------
extern "C" void kernel_launch(void* const* d_in, const int* in_sizes, int n_in,
                              void* d_out, int out_size, void* d_ws, size_t ws_size,
                              hipStream_t stream)
{
    const float* x           = (const float*)d_in[0];
    const int*   edge_index  = (const int*)d_in[1];
    const float* edge_attr   = (const float*)d_in[2];
    const int*   valid_act   = (const int*)d_in[3];
    const int*   current_nd  = (const int*)d_in[4];
    const int*   cur_path    = (const int*)d_in[5];
    const int*   coll_paths  = (const int*)d_in[6];
    const float* W_node      = (const float*)d_in[7];
    const float* b_node      = (const float*)d_in[8];
    const float* W_eenc      = (const float*)d_in[9];
    const float* b_eenc      = (const float*)d_in[10];
    const float* conv_lin    = (const float*)d_in[11];
    const float* att_src     = (const float*)d_in[12];
    const float* att_dst     = (const float*)d_in[13];
    const float* conv_lin_e  = (const float*)d_in[14];
    const float* att_edge    = (const float*)d_in[15];
    const float* conv_bias   = (const float*)d_in[16];
    const float* pos_emb     = (const float*)d_in[17];
    const float* W_comb      = (const float*)d_in[18];
    const float* b_comb      = (const float*)d_in[19];
    const float* W_s1        = (const float*)d_in[20];
    const float* b_s1        = (const float*)d_in[21];
    const float* W_s2        = (const float*)d_in[22];
    const float* b_s2        = (const float*)d_in[23];
    float*       out         = (float*)d_out;

    const int N = in_sizes[0] / NF;     // 50000
    const int E = in_sizes[2] / FE;     // 800000
    const int* src = edge_index;
    const int* dst = edge_index + E;

    // ---- workspace carve (floats) ----
    float* w        = (float*)d_ws;
    float* acc      = w;              w += (long)N * 64;
    float* denom    = w;              w += N;
    float* sum_ale  = w;              w += N;   // acc|denom|sum_ale contiguous -> one zero pass
    float* cnt      = w;              w += N;
    float* al_s     = w;              w += N;
    float* al_d     = w;              w += N;
    float* loopale  = w;              w += N;
    float* hbuf     = w;              w += (long)N * 64;
    float* xs       = w;              w += (long)N * 64;
    float* al_e     = w;              w += E;
    float* uc       = w;              w += 32;
    float* pemb     = w;              w += 64;
    float* cemb     = w;              w += 64;

    const int ZT = 256;
    const int gridGemm = (N + 31) / 32;
    const int gridN    = (N + ZT - 1) / ZT;
    const int gridE    = (E + ZT - 1) / ZT;
    const int gridAgg  = (E + N + 7) / 8;        // 8 edges (waves) per block
    const int gridNF   = (N * 64 + ZT - 1) / ZT;

    // in-degree counts (once)
    fill_zero_kernel<<<gridN, ZT, 0, stream>>>(cnt, N);
    count_kernel<<<gridE, ZT, 0, stream>>>(dst, cnt, E);

    // node encoder: h = x @ W_node + b_node  (WMMA)
    gemm_n64_k64_wmma<<<gridGemm, 256, 0, stream>>>(x, W_node, b_node, hbuf, N);

    for (int l = 0; l < 2; ++l) {
        const float* lin   = conv_lin   + (long)l * 64 * 64;
        const float* lin_e = conv_lin_e + (long)l * 64 * 64;
        const float* a_s   = att_src  + l * 64;
        const float* a_d   = att_dst  + l * 64;
        const float* a_e   = att_edge + l * 64;
        const float* bi    = conv_bias + l * 64;

        // fold edge-feature path to a 16-vector + constant
        ve_u_kernel<<<1, 64, 0, stream>>>(lin_e, a_e, W_eenc, b_eenc, uc);

        // xs = h @ lin  (WMMA)
        gemm_n64_k64_wmma<<<gridGemm, 256, 0, stream>>>(hbuf, lin, nullptr, xs, N);

        // attention logit pieces
        alsd_kernel<<<gridN, ZT, 0, stream>>>(xs, a_s, a_d, al_s, al_d, N);

        // zero acc + denom + sum_ale (contiguous)
        fill_zero_kernel<<<(N * 64 + 2 * N + ZT - 1) / ZT, ZT, 0, stream>>>(acc, N * 64 + 2 * N);

        // per-edge al_e + per-dst sum for self-loop attr
        ale_kernel<<<gridE, ZT, 0, stream>>>(edge_attr, dst, uc, al_e, sum_ale, E);
        loopale_kernel<<<gridN, ZT, 0, stream>>>(sum_ale, cnt, loopale, N);

        // softmax-weighted scatter aggregation (real + self-loop edges)
        edge_agg_kernel<<<gridAgg, 256, 0, stream>>>(src, dst, al_s, al_d, al_e, loopale,
                                                     xs, acc, denom, E, N);

        // h = relu(acc/denom + bias)
        finalize_kernel<<<gridNF, ZT, 0, stream>>>(acc, denom, bi, hbuf, N);
    }

    // path encodings + scoring
    path_kernel<<<1, 64, 0, stream>>>(hbuf, cur_path, coll_paths, pos_emb, W_comb, b_comb,
                                      pemb, cemb);
    score_kernel<<<1, 256, 0, stream>>>(hbuf, valid_act, current_nd, pemb, cemb,
                                        W_s1, b_s1, W_s2, b_s2, out);
}